// NerfRenderer_59648505806896
// MI455X (gfx1250) — compile-verified
//
#include <hip/hip_runtime.h>
#include <hip/hip_bf16.h>
#include <stdint.h>

// ---------------------------------------------------------------------------
// NeRF occupancy grid update for MI455X (gfx1250).
// Memory-bound scatter/pack workload: no matmul -> no WMMA; instead we use the
// CDNA5 async global->LDS data movers (GLOBAL_LOAD_ASYNC_TO_LDS_B128 +
// S_WAIT_ASYNCCNT) to turn strided coord reads into bulk B128 DMA and to
// software-pipeline the packbits pass.
// ---------------------------------------------------------------------------

#define GRID_N   (1 << 24)          // 256^3 cells per cascade
#define CASCADES 3
#define NCOORD   (GRID_N / 4)       // 4,194,304 samples per cascade (== 1<<22)
#define CG       (CASCADES * GRID_N)
#define CN       (CASCADES * NCOORD)
#define DECAY          0.95f
#define DENSITY_SCALE  1.0f
#define THRESH         0.01f

// ---- CDNA5 async global -> LDS copy (ASYNCcnt-tracked) --------------------
__device__ __forceinline__ void async_ld_b128(unsigned lds_byte_off, const void* gptr) {
    asm volatile("global_load_async_to_lds_b128 %0, %1, off"
                 :
                 : "v"(lds_byte_off), "v"((unsigned long long)(uintptr_t)gptr)
                 : "memory");
}

template <int N>
__device__ __forceinline__ void wait_asynccnt() {
    asm volatile("s_wait_asynccnt %0" : : "i"(N) : "memory");
}

// ---- Morton (Z-order) encode ----------------------------------------------
__device__ __forceinline__ unsigned part1by2(unsigned x) {
    x &= 0x3FFu;
    x = (x | (x << 16)) & 0x030000FFu;
    x = (x | (x << 8))  & 0x0300F00Fu;
    x = (x | (x << 4))  & 0x030C30C3u;
    x = (x | (x << 2))  & 0x09249249u;
    return x;
}

// ---------------------------------------------------------------------------
// Kernel 1: scatter-max.  One thread per (cascade, sample).
// Block of 256 threads async-DMAs its contiguous 3 KiB coord slab (256 x int3)
// into LDS as 192 B128 transactions, then each thread Morton-encodes and does
// the decay/max combine directly into the output grid (base copy done by a
// prior D2D memcpy on the stream).  Duplicate-index write races match the
// reference's nondeterministic scatter-set semantics.
// ---------------------------------------------------------------------------
__global__ __launch_bounds__(256) void scatter_max_kernel(
    const float* __restrict__ density,
    const float* __restrict__ sigmas,
    const int*   __restrict__ coords,
    float*       __restrict__ grid_out)
{
    __shared__ int sc[256 * 3];                 // 3 KiB staged coords
    const int tid = threadIdx.x;
    const long long t0 = (long long)blockIdx.x * 256;

    // 3072 bytes = 192 lanes x 16B async copies
    if (tid < 192) {
        const char* g = (const char*)coords + (size_t)t0 * 12 + (size_t)tid * 16;
        unsigned l = (unsigned)(uintptr_t)sc + (unsigned)(tid * 16);
        async_ld_b128(l, g);
    }
    wait_asynccnt<0>();      // per-wave: my async copies have landed in LDS
    __syncthreads();         // cross-wave visibility of the staged slab

    const int t = (int)(t0 + tid);              // CN < 2^31
    if (t < CN) {
        const unsigned cx = (unsigned)sc[tid * 3 + 0];
        const unsigned cy = (unsigned)sc[tid * 3 + 1];
        const unsigned cz = (unsigned)sc[tid * 3 + 2];
        const unsigned m  = part1by2(cx) | (part1by2(cy) << 1) | (part1by2(cz) << 2);

        const int cascade = t >> 22;            // t / NCOORD
        const size_t gi = ((size_t)cascade << 24) | (size_t)m;

        const float s = sigmas[t] * DENSITY_SCALE;
        const float d = density[gi];
        if (d >= 0.0f && s >= 0.0f) {
            grid_out[gi] = fmaxf(d * DECAY, s); // L2-resident RMW (192MB L2)
        }
    }
}

// ---------------------------------------------------------------------------
// Kernel 2: packbits.  Each output value consumes 8 consecutive grid floats.
// Double-buffered async pipeline: while packing tile k from LDS, tile k+stride
// is in flight (s_wait_asynccnt 2 leaves the next tile's two B128 loads
// outstanding).  Each lane owns its private 32B slot per buffer -> no barrier
// needed after the per-wave asynccnt wait.
// ---------------------------------------------------------------------------
#define PB_TILE 256   // outputs per tile; 256 threads x 32B = 8 KiB per buffer

__global__ __launch_bounds__(256) void packbits_kernel(
    const float* __restrict__ grid,
    float*       __restrict__ bits)
{
    __shared__ float4 buf[2][PB_TILE * 2];      // 2 x 8 KiB double buffer
    const int tid      = threadIdx.x;
    const int numTiles = (CG / 8) / PB_TILE;    // 24576
    const int stride   = gridDim.x;
    int tile = blockIdx.x;
    if (tile >= numTiles) return;

    const unsigned lbase = (unsigned)(uintptr_t)&buf[0][0];

    auto issue = [&](int t, int b) {
        const float* g = grid + (size_t)t * (PB_TILE * 8) + (size_t)tid * 8;
        const unsigned l = lbase + (unsigned)(b * (PB_TILE * 2 * 16))
                                 + (unsigned)(tid * 32);
        async_ld_b128(l,      g);
        async_ld_b128(l + 16, g + 4);
    };

    int b = 0;
    issue(tile, b);                             // prologue: fill buffer 0

    for (; tile < numTiles; tile += stride) {
        const int  nxt     = tile + stride;
        const bool hasNext = nxt < numTiles;
        if (hasNext) {
            issue(nxt, b ^ 1);                  // keep next tile in flight
            wait_asynccnt<2>();                 // current tile's 2 loads done
        } else {
            wait_asynccnt<0>();
        }

        const float4 lo = buf[b][tid * 2 + 0];  // ds_load_b128
        const float4 hi = buf[b][tid * 2 + 1];

        unsigned m = 0;
        m |= (lo.x > THRESH) ?   1u : 0u;
        m |= (lo.y > THRESH) ?   2u : 0u;
        m |= (lo.z > THRESH) ?   4u : 0u;
        m |= (lo.w > THRESH) ?   8u : 0u;
        m |= (hi.x > THRESH) ?  16u : 0u;
        m |= (hi.y > THRESH) ?  32u : 0u;
        m |= (hi.z > THRESH) ?  64u : 0u;
        m |= (hi.w > THRESH) ? 128u : 0u;

        bits[(size_t)tile * PB_TILE + tid] = (float)m;
        b ^= 1;
    }
}

// ---------------------------------------------------------------------------
extern "C" void kernel_launch(void* const* d_in, const int* in_sizes, int n_in,
                              void* d_out, int out_size, void* d_ws, size_t ws_size,
                              hipStream_t stream) {
    (void)in_sizes; (void)n_in; (void)out_size; (void)d_ws; (void)ws_size;

    const float* density = (const float*)d_in[0];   // [C, G]
    const float* sigmas  = (const float*)d_in[1];   // [C, N]
    const int*   coords  = (const int*)  d_in[2];   // [C, N, 3]

    float* out_grid = (float*)d_out;                // [C*G] new_grid
    float* out_bits = out_grid + (size_t)CG;        // [C*G/8] bitfield (as float)

    // Pass 1: new_grid = density_grid (cells untouched by the scatter keep it).
    hipMemcpyAsync(out_grid, density, (size_t)CG * sizeof(float),
                   hipMemcpyDeviceToDevice, stream);

    // Pass 2: Morton + scatter decay/max (async-LDS staged coords).
    scatter_max_kernel<<<CN / 256, 256, 0, stream>>>(density, sigmas, coords, out_grid);

    // Pass 3: packbits (async double-buffered pipeline).
    packbits_kernel<<<4096, 256, 0, stream>>>(out_grid, out_bits);
}